// SMPL_37203006718137
// MI455X (gfx1250) — compile-verified
//
#include <hip/hip_runtime.h>
#include <math.h>

// ---------------- problem constants ----------------
#define NV     6890
#define NV3    20670      // NV*3
#define NBETA  10
#define NJNT   24
#define NPOSE  207
#define NKOUT  19
#define KTOT   220        // 10 beta + 2 pad + 207 pose + 1 pad  (multiple of 4)
#define KCHUNK 20         // K rows staged per async chunk
#define NCHUNK (KTOT / KCHUNK)   // 11

// fused kernel tiling
#define TPB        192    // 6 waves
#define WAVES      6
#define COLS_BLK   96     // 6 waves * 16 cols = 32 whole vertices
#define VERTS_BLK  32

typedef float v2f __attribute__((ext_vector_type(2)));
typedef float v8f __attribute__((ext_vector_type(8)));

__constant__ int c_parents[NJNT] = {-1, 0, 0, 0, 1, 2, 3, 4, 5, 6, 7, 8,
                                    9, 9, 9, 12, 13, 14, 16, 17, 18, 19, 20, 21};

__device__ __forceinline__ int imin(int a, int b) { return a < b ? a : b; }

// ---------------------------------------------------------------------------
// Kernel 1: Rodrigues -> Rs (B,24,9) and pose_feature (B,207)
// ---------------------------------------------------------------------------
__global__ void rodrigues_kernel(const float* __restrict__ theta,
                                 float* __restrict__ Rs,
                                 float* __restrict__ pf, int B) {
    int idx = blockIdx.x * blockDim.x + threadIdx.x;
    if (idx >= B * NJNT) return;
    int b = idx / NJNT, j = idx % NJNT;
    float tx = theta[b * (NJNT * 3) + j * 3 + 0];
    float ty = theta[b * (NJNT * 3) + j * 3 + 1];
    float tz = theta[b * (NJNT * 3) + j * 3 + 2];
    float ax = tx + 1e-8f, ay = ty + 1e-8f, az = tz + 1e-8f;
    float angle = sqrtf(ax * ax + ay * ay + az * az);
    float half = 0.5f * angle;
    float s = sinf(half) / angle;
    float qw = cosf(half), qx = tx * s, qy = ty * s, qz = tz * s;
    float qn = rsqrtf(qw * qw + qx * qx + qy * qy + qz * qz);
    qw *= qn; qx *= qn; qy *= qn; qz *= qn;
    float rot[9];
    rot[0] = 1.f - 2.f * (qy * qy + qz * qz);
    rot[1] = 2.f * (qx * qy - qw * qz);
    rot[2] = 2.f * (qx * qz + qw * qy);
    rot[3] = 2.f * (qx * qy + qw * qz);
    rot[4] = 1.f - 2.f * (qx * qx + qz * qz);
    rot[5] = 2.f * (qy * qz - qw * qx);
    rot[6] = 2.f * (qx * qz - qw * qy);
    rot[7] = 2.f * (qy * qz + qw * qx);
    rot[8] = 1.f - 2.f * (qx * qx + qy * qy);
    float* rs = Rs + (size_t)b * (NJNT * 9) + j * 9;
    #pragma unroll
    for (int e = 0; e < 9; ++e) rs[e] = rot[e];
    if (j > 0) {
        float* p = pf + (size_t)b * NPOSE + (j - 1) * 9;
        #pragma unroll
        for (int e = 0; e < 9; ++e)
            p[e] = rot[e] - ((e == 0 || e == 4 || e == 8) ? 1.f : 0.f);
    }
}

// ---------------------------------------------------------------------------
// Kernel 2: Jt[24][3] = J_regressor^T . v_template ;
//           Sjr[10][24][3] = J_regressor^T . shapedirs (reshaped)
// ---------------------------------------------------------------------------
__global__ void regress_consts_kernel(const float* __restrict__ v_template,
                                      const float* __restrict__ shapedirs,
                                      const float* __restrict__ Jreg,
                                      float* __restrict__ Jt,
                                      float* __restrict__ Sjr) {
    int j = blockIdx.x / 3, c = blockIdx.x % 3;
    __shared__ float red[256];
    float acc[1 + NBETA];
    #pragma unroll
    for (int t = 0; t < 1 + NBETA; ++t) acc[t] = 0.f;
    for (int v = threadIdx.x; v < NV; v += 256) {
        float jw = Jreg[v * NJNT + j];
        acc[0] += v_template[v * 3 + c] * jw;
        #pragma unroll
        for (int nb = 0; nb < NBETA; ++nb)
            acc[1 + nb] += shapedirs[(size_t)nb * NV3 + v * 3 + c] * jw;
    }
    for (int t = 0; t < 1 + NBETA; ++t) {
        red[threadIdx.x] = acc[t];
        __syncthreads();
        for (int s = 128; s > 0; s >>= 1) {
            if (threadIdx.x < s) red[threadIdx.x] += red[threadIdx.x + s];
            __syncthreads();
        }
        if (threadIdx.x == 0) {
            if (t == 0) Jt[j * 3 + c] = red[0];
            else        Sjr[((t - 1) * NJNT + j) * 3 + c] = red[0];
        }
        __syncthreads();
    }
}

// ---------------------------------------------------------------------------
// Kernel 3: kinematic chain per body -> A (B,24,12)
// ---------------------------------------------------------------------------
__global__ void chain_kernel(const float* __restrict__ beta,
                             const float* __restrict__ Rs,
                             const float* __restrict__ Jt,
                             const float* __restrict__ Sjr,
                             float* __restrict__ Aout, int B) {
    int b = blockIdx.x * blockDim.x + threadIdx.x;
    if (b >= B) return;
    float J[NJNT][3];
    #pragma unroll 4
    for (int j = 0; j < NJNT; ++j)
        for (int c = 0; c < 3; ++c) {
            float v = Jt[j * 3 + c];
            for (int nb = 0; nb < NBETA; ++nb)
                v += beta[b * NBETA + nb] * Sjr[(nb * NJNT + j) * 3 + c];
            J[j][c] = v;
        }
    float R[NJNT][9], T[NJNT][3];
    const float* Rb = Rs + (size_t)b * (NJNT * 9);
    for (int r = 0; r < 3; ++r) {
        R[0][r * 3 + 0] =  Rb[r * 3 + 0];
        R[0][r * 3 + 1] = -Rb[r * 3 + 1];
        R[0][r * 3 + 2] = -Rb[r * 3 + 2];
        T[0][r] = J[0][r];
    }
    for (int i = 1; i < NJNT; ++i) {
        int p = c_parents[i];
        float t0 = J[i][0] - J[p][0], t1 = J[i][1] - J[p][1], t2 = J[i][2] - J[p][2];
        const float* Ri = Rb + i * 9;
        for (int r = 0; r < 3; ++r) {
            float p0 = R[p][r * 3 + 0], p1 = R[p][r * 3 + 1], p2 = R[p][r * 3 + 2];
            for (int c = 0; c < 3; ++c)
                R[i][r * 3 + c] = p0 * Ri[0 * 3 + c] + p1 * Ri[1 * 3 + c] + p2 * Ri[2 * 3 + c];
            T[i][r] = p0 * t0 + p1 * t1 + p2 * t2 + T[p][r];
        }
    }
    float* out = Aout + (size_t)b * (NJNT * 12);
    for (int j = 0; j < NJNT; ++j)
        for (int r = 0; r < 3; ++r) {
            float r0 = R[j][r * 3 + 0], r1 = R[j][r * 3 + 1], r2 = R[j][r * 3 + 2];
            out[j * 12 + r * 4 + 0] = r0;
            out[j * 12 + r * 4 + 1] = r1;
            out[j * 12 + r * 4 + 2] = r2;
            out[j * 12 + r * 4 + 3] =
                T[j][r] - (r0 * J[j][0] + r1 * J[j][1] + r2 * J[j][2]);
        }
}

// ---------------------------------------------------------------------------
// Kernel 4: zero output accumulators
// ---------------------------------------------------------------------------
__global__ void zero_kernel(float* __restrict__ out, int n) {
    int i = blockIdx.x * blockDim.x + threadIdx.x;
    if (i < n) out[i] = 0.f;
}

// ---------------------------------------------------------------------------
// Kernel 5 (fused): async-LDS double-buffered WMMA fp32 GEMM -> LBS skinning
// -> joint regression. Inner loop is pure LDS + WMMA; B operand chunks stream
// in via GLOBAL_LOAD_ASYNC_TO_LDS_B64 (ASYNCcnt) overlapped with compute.
// ---------------------------------------------------------------------------
__device__ __forceinline__ const float* brow(const float* __restrict__ sdirs,
                                             const float* __restrict__ pdirs,
                                             int k) {
    // combined K layout: [0,10) shapedirs, [10,12) pad, [12,219) posedirs, 219 pad
    if (k < 12) return sdirs + (size_t)imin(k, NBETA - 1) * NV3;
    return pdirs + (size_t)imin(k - 12, NPOSE - 1) * NV3;
}

__global__ __launch_bounds__(TPB)
void fused_skin_kernel(const float* __restrict__ beta,
                       const float* __restrict__ pf,
                       const float* __restrict__ shapedirs,
                       const float* __restrict__ posedirs,
                       const float* __restrict__ v_template,
                       const float* __restrict__ Amat,
                       const float* __restrict__ weights,
                       const float* __restrict__ jreg,
                       const float* __restrict__ trans,
                       float* __restrict__ out) {
    __shared__ float lds_pfA[16 * KTOT];                    // A operand (16 x 220)
    __shared__ __align__(16) float lds_B[2][KCHUNK * COLS_BLK]; // B double buffer
    __shared__ float lds_A[16 * NJNT * 12];                 // bone matrices
    __shared__ float lds_vp[16 * COLS_BLK];                 // v_posed tile
    __shared__ float lds_w[VERTS_BLK * NJNT];               // skinning weights
    __shared__ float lds_jr[VERTS_BLK * NKOUT];             // joint regressor slice
    __shared__ float lds_jp[16 * NKOUT * 3];                // joint partials

    const int tid   = threadIdx.x;
    const int b0    = blockIdx.y * 16;
    const int cblk  = blockIdx.x * COLS_BLK;
    const int vbase = blockIdx.x * VERTS_BLK;

    // Async-stage one K-chunk (20 rows x 96 cols) of the combined B matrix
    // into LDS buffer `buf` with b64 transfers (row stride is 8B-aligned).
    auto stage_chunk = [&](int chunk, int buf) {
        #pragma unroll
        for (int i = 0; i < (KCHUNK * COLS_BLK / 2) / TPB; ++i) {   // 5 iters
            int q   = tid + i * TPB;           // 0..959 (b64 units)
            int row = q / (COLS_BLK / 2);      // 0..19
            int e   = q % (COLS_BLK / 2);      // 0..47
            int gcol = imin(cblk + e * 2, NV3 - 2);
            const float* src = brow(shapedirs, posedirs, chunk * KCHUNK + row) + gcol;
            unsigned lds_off =
                (unsigned)(uintptr_t)&lds_B[buf][row * COLS_BLK + e * 2];
            asm volatile("global_load_async_to_lds_b64 %0, %1, off"
                         :: "v"(lds_off), "v"(src)
                         : "memory");
            if (i == 0 && chunk + 1 < NCHUNK) {   // warm L2 one chunk ahead
                const float* pre =
                    brow(shapedirs, posedirs,
                         imin(chunk * KCHUNK + row + KCHUNK, KTOT - 1)) + gcol;
                __builtin_prefetch(pre, 0, 1);
            }
        }
    };

    // ---- stage combined A operand (beta | 0 | pose_feature | 0) ----
    for (int idx = tid; idx < 16 * KTOT; idx += TPB) {
        int row = idx / KTOT, k = idx % KTOT;
        float val = 0.f;
        if (k < NBETA)            val = beta[(size_t)(b0 + row) * NBETA + k];
        else if (k >= 12 && k < 12 + NPOSE)
                                  val = pf[(size_t)(b0 + row) * NPOSE + (k - 12)];
        lds_pfA[idx] = val;
    }
    // ---- stage bone matrices, weights, regressor, zero partials ----
    for (int idx = tid; idx < 16 * NJNT * 12; idx += TPB)
        lds_A[idx] = Amat[(size_t)b0 * (NJNT * 12) + idx];
    for (int idx = tid; idx < VERTS_BLK * NJNT; idx += TPB) {
        int vl = idx / NJNT, j = idx % NJNT;
        int v = imin(vbase + vl, NV - 1);
        lds_w[idx] = weights[(size_t)v * NJNT + j];
    }
    for (int idx = tid; idx < VERTS_BLK * NKOUT; idx += TPB) {
        int vl = idx / NKOUT, k = idx % NKOUT;
        int v = vbase + vl;
        lds_jr[idx] = (v < NV) ? jreg[(size_t)v * NKOUT + k] : 0.f;
    }
    for (int idx = tid; idx < 16 * NKOUT * 3; idx += TPB) lds_jp[idx] = 0.f;

    // kick off first B chunk while the LDS stores above drain
    stage_chunk(0, 0);
    __syncthreads();

    // ---- WMMA fp32 GEMM: 16(batch) x 16(col) tile per wave, K = 220 ----
    const int wave  = tid >> 5;
    const int lane  = tid & 31;
    const int m     = lane & 15;          // A row / B col within tile
    const int khalf = (lane >> 4) * 2;    // K sub-lane split per ISA layout
    const int wcol  = wave * 16 + m;      // column within block (0..95)
    const int col   = imin(cblk + wcol, NV3 - 1);

    v8f acc = {};
    for (int c = 0; c < NCHUNK; ++c) {
        asm volatile("s_wait_asynccnt 0x0" ::: "memory");  // chunk c resident
        __syncthreads();
        if (c + 1 < NCHUNK) stage_chunk(c + 1, (c + 1) & 1);
        const float* Bb = &lds_B[c & 1][0];
        #pragma unroll
        for (int kk = 0; kk < KCHUNK; kk += 4) {
            int kb = kk + khalf;
            v2f a, bv;
            a.x  = lds_pfA[m * KTOT + c * KCHUNK + kb];
            a.y  = lds_pfA[m * KTOT + c * KCHUNK + kb + 1];
            bv.x = Bb[kb * COLS_BLK + wcol];
            bv.y = Bb[(kb + 1) * COLS_BLK + wcol];
            acc = __builtin_amdgcn_wmma_f32_16x16x4_f32(
                false, a, false, bv, (short)0, acc, false, false);
        }
    }

    // ---- add v_template (per-column constant) and spill tile to LDS ----
    {
        float vt = v_template[col];
        int rbase = (lane >> 4) * 8;               // C/D layout: row = i + 8*(lane>>4)
        #pragma unroll
        for (int i = 0; i < 8; ++i)
            lds_vp[(rbase + i) * COLS_BLK + wcol] = acc[i] + vt;
    }
    __syncthreads();

    // ---- LBS skinning + joint regression partials ----
    for (int pair = tid; pair < 16 * VERTS_BLK; pair += TPB) {
        int b  = pair >> 5;        // batch row in tile
        int vl = pair & 31;        // vertex in tile
        int v  = vbase + vl;
        if (v < NV) {
            float p0 = lds_vp[b * COLS_BLK + vl * 3 + 0];
            float p1 = lds_vp[b * COLS_BLK + vl * 3 + 1];
            float p2 = lds_vp[b * COLS_BLK + vl * 3 + 2];
            float vx = 0.f, vy = 0.f, vz = 0.f;
            #pragma unroll 4
            for (int j = 0; j < NJNT; ++j) {
                float w = lds_w[vl * NJNT + j];
                const float* Aj = &lds_A[b * (NJNT * 12) + j * 12];
                vx += w * (Aj[0] * p0 + Aj[1] * p1 + Aj[2]  * p2 + Aj[3]);
                vy += w * (Aj[4] * p0 + Aj[5] * p1 + Aj[6]  * p2 + Aj[7]);
                vz += w * (Aj[8] * p0 + Aj[9] * p1 + Aj[10] * p2 + Aj[11]);
            }
            vx += trans[(b0 + b) * 3 + 0];
            vy += trans[(b0 + b) * 3 + 1];
            vz += trans[(b0 + b) * 3 + 2];
            #pragma unroll
            for (int k = 0; k < NKOUT; ++k) {
                float jv = lds_jr[vl * NKOUT + k];
                atomicAdd(&lds_jp[(b * NKOUT + k) * 3 + 0], jv * vx);
                atomicAdd(&lds_jp[(b * NKOUT + k) * 3 + 1], jv * vy);
                atomicAdd(&lds_jp[(b * NKOUT + k) * 3 + 2], jv * vz);
            }
        }
    }
    __syncthreads();

    // ---- reduce block partials into global joints ----
    for (int idx = tid; idx < 16 * NKOUT * 3; idx += TPB) {
        int b = idx / (NKOUT * 3);
        int rem = idx % (NKOUT * 3);
        atomicAdd(&out[(size_t)(b0 + b) * (NKOUT * 3) + rem], lds_jp[idx]);
    }
}

// ---------------------------------------------------------------------------
extern "C" void kernel_launch(void* const* d_in, const int* in_sizes, int n_in,
                              void* d_out, int out_size, void* d_ws, size_t ws_size,
                              hipStream_t stream) {
    const float* beta        = (const float*)d_in[0];
    const float* theta       = (const float*)d_in[1];
    const float* trans       = (const float*)d_in[2];
    const float* v_template  = (const float*)d_in[3];
    const float* shapedirs   = (const float*)d_in[4];
    const float* J_regressor = (const float*)d_in[5];
    const float* posedirs    = (const float*)d_in[6];
    const float* joint_reg   = (const float*)d_in[7];
    const float* weights     = (const float*)d_in[8];
    float* out = (float*)d_out;
    const int B = in_sizes[0] / NBETA;

    float* ws = (float*)d_ws;
    float* pfb  = ws;                 size_t o = (size_t)B * NPOSE;
    float* Rs   = ws + o;             o += (size_t)B * NJNT * 9;
    float* Amat = ws + o;             o += (size_t)B * NJNT * 12;
    float* Jt   = ws + o;             o += NJNT * 3;
    float* Sjr  = ws + o;             o += NBETA * NJNT * 3;

    rodrigues_kernel<<<(B * NJNT + 255) / 256, 256, 0, stream>>>(theta, Rs, pfb, B);
    regress_consts_kernel<<<NJNT * 3, 256, 0, stream>>>(v_template, shapedirs,
                                                        J_regressor, Jt, Sjr);
    chain_kernel<<<(B + 63) / 64, 64, 0, stream>>>(beta, Rs, Jt, Sjr, Amat, B);
    zero_kernel<<<(B * NKOUT * 3 + 255) / 256, 256, 0, stream>>>(out, B * NKOUT * 3);

    dim3 grid((NV3 + COLS_BLK - 1) / COLS_BLK, B / 16);
    fused_skin_kernel<<<grid, TPB, 0, stream>>>(beta, pfb, shapedirs, posedirs,
                                                v_template, Amat, weights,
                                                joint_reg, trans, out);
}